// EGNN_55611236548999
// MI455X (gfx1250) — compile-verified
//
#include <hip/hip_runtime.h>

typedef __attribute__((ext_vector_type(2))) float v2f;
typedef __attribute__((ext_vector_type(8))) float v8f;

#define HID 16
#define RRELU_SLOPE ((1.0f / 8.0f + 1.0f / 3.0f) * 0.5f)

// ---------------------------------------------------------------------------
// Degree / normalization
// ---------------------------------------------------------------------------
__global__ __launch_bounds__(256) void k_init_deg(float* __restrict__ deg, int n) {
    int i = blockIdx.x * blockDim.x + threadIdx.x;
    if (i < n) deg[i] = 1.0f;  // self loop contributes 1 to every degree
}

__global__ __launch_bounds__(256) void k_deg_accum(const int* __restrict__ col,
                                                   float* __restrict__ deg, int E) {
    int e = blockIdx.x * blockDim.x + threadIdx.x;
    if (e < E) atomicAdd(&deg[col[e]], 1.0f);
}

__global__ __launch_bounds__(256) void k_deg_to_dis(float* __restrict__ deg, int n) {
    int i = blockIdx.x * blockDim.x + threadIdx.x;
    if (i < n) deg[i] = 1.0f / sqrtf(deg[i]);  // deg >= 1 always (self loop)
}

// ---------------------------------------------------------------------------
// Dense transform: xw[n,16] = act(A[n,kdim]) @ W[kdim,16] via V_WMMA_F32_16X16X4_F32
// One wave handles one 16-row tile. K processed in chunks of 4 (f32 WMMA shape).
//
// f32 A (16x4) layout: lanes 0-15 -> M=lane, VGPR{0,1} = K={0,1};
//                      lanes 16-31 -> M=lane-16, VGPR{0,1} = K={2,3}.
// f32 B (4x16) layout mirrors A (N across lanes, K split by lane half).
// f32 C/D (16x16): VGPR r -> row r (lanes 0-15) / row 8+r (lanes 16-31), N=lane&15.
// ---------------------------------------------------------------------------
__global__ __launch_bounds__(256) void k_dense_wmma(const float* __restrict__ A, int lda,
                                                    int kdim,
                                                    const float* __restrict__ W,
                                                    float* __restrict__ xw,
                                                    int n, int relu_in) {
    int wave = (int)((blockIdx.x * blockDim.x + threadIdx.x) >> 5);
    int lane = threadIdx.x & 31;
    int base = wave * 16;
    if (base >= n) return;            // uniform per wave: EXEC stays all-ones for WMMA

    int half = lane >> 4;             // 0: K lo pair, 1: K hi pair
    int nidx = lane & 15;             // A row within tile / B+D column
    int arow = base + nidx;
    if (arow >= n) arow = n - 1;      // clamp tail-tile loads (stores guarded below)

    v8f c = {0.f, 0.f, 0.f, 0.f, 0.f, 0.f, 0.f, 0.f};
    int nchunks = kdim >> 2;
    for (int kc = 0; kc < nchunks; ++kc) {
        int k0 = kc * 4 + half * 2;
        float a0 = A[arow * lda + k0];
        float a1 = A[arow * lda + k0 + 1];
        if (relu_in) { a0 = fmaxf(a0, 0.f); a1 = fmaxf(a1, 0.f); }
        v2f a; a.x = a0; a.y = a1;
        v2f b; b.x = W[(k0) * HID + nidx]; b.y = W[(k0 + 1) * HID + nidx];
        c = __builtin_amdgcn_wmma_f32_16x16x4_f32(false, a, false, b,
                                                  (short)0, c, false, false);
    }
#pragma unroll
    for (int r = 0; r < 8; ++r) {
        int row = base + half * 8 + r;
        if (row < n) xw[row * HID + nidx] = c[r];
    }
}

// ---------------------------------------------------------------------------
// acc[i,:] = bias + dis[i]^2 * xw[i,:]   (self-loop term folded into init)
// ---------------------------------------------------------------------------
__global__ __launch_bounds__(256) void k_init_acc(const float* __restrict__ dis,
                                                  const float* __restrict__ xw,
                                                  const float* __restrict__ bias,
                                                  float* __restrict__ acc, int n) {
    int i = blockIdx.x * blockDim.x + threadIdx.x;
    if (i >= n) return;
    float s = dis[i];
    s = s * s;
    const float4* src = (const float4*)(xw + (size_t)i * HID);
    const float4* bv  = (const float4*)bias;
    float4* dst       = (float4*)(acc + (size_t)i * HID);
#pragma unroll
    for (int j = 0; j < 4; ++j) {
        float4 v = src[j];
        float4 b = bv[j];
        dst[j] = make_float4(b.x + s * v.x, b.y + s * v.y,
                             b.z + s * v.z, b.w + s * v.w);
    }
}

// ---------------------------------------------------------------------------
// Edge scatter: acc[col,:] += dis[row]*dis[col] * xw[row,:]
// ---------------------------------------------------------------------------
__global__ __launch_bounds__(256) void k_edge_scatter(const int* __restrict__ row,
                                                      const int* __restrict__ colv,
                                                      const float* __restrict__ dis,
                                                      const float* __restrict__ xw,
                                                      float* __restrict__ acc, int E) {
    int e = blockIdx.x * blockDim.x + threadIdx.x;
    if (e >= E) return;
    int r = row[e];
    int c = colv[e];
    float nrm = dis[r] * dis[c];
    const float4* src = (const float4*)(xw + (size_t)r * HID);
    float* dst = acc + (size_t)c * HID;
#pragma unroll
    for (int j = 0; j < 4; ++j) {
        float4 v = src[j];
        atomicAdd(dst + 4 * j + 0, nrm * v.x);
        atomicAdd(dst + 4 * j + 1, nrm * v.y);
        atomicAdd(dst + 4 * j + 2, nrm * v.z);
        atomicAdd(dst + 4 * j + 3, nrm * v.w);
    }
}

// ---------------------------------------------------------------------------
// Pooling + MLP head
// ---------------------------------------------------------------------------
__global__ __launch_bounds__(256) void k_zero(float* __restrict__ p, int n) {
    int i = blockIdx.x * blockDim.x + threadIdx.x;
    if (i < n) p[i] = 0.0f;
}

__global__ __launch_bounds__(256) void k_pool(const float* __restrict__ acc,
                                              const int* __restrict__ batch,
                                              float* __restrict__ gsum,
                                              float* __restrict__ gcnt, int n) {
    int i = blockIdx.x * blockDim.x + threadIdx.x;
    if (i >= n) return;
    int g = batch[i];
    const float* h = acc + (size_t)i * HID;
    float* d = gsum + (size_t)g * HID;
#pragma unroll
    for (int f = 0; f < HID; ++f) atomicAdd(d + f, fmaxf(h[f], 0.0f));  // final relu fused
    atomicAdd(gcnt + g, 1.0f);
}

__global__ __launch_bounds__(256) void k_mlp(const float* __restrict__ gsum,
                                             const float* __restrict__ gcnt,
                                             const float* __restrict__ w1,
                                             const float* __restrict__ b1,
                                             const float* __restrict__ w2,
                                             const float* __restrict__ b2,
                                             float* __restrict__ out, int G) {
    int i = blockIdx.x * blockDim.x + threadIdx.x;
    if (i >= G) return;
    float inv = 1.0f / fmaxf(gcnt[i], 1.0f);
    float g[HID];
#pragma unroll
    for (int k = 0; k < HID; ++k) g[k] = gsum[(size_t)i * HID + k] * inv;
    float z = b2[0];
#pragma unroll
    for (int j = 0; j < HID; ++j) {
        float a = b1[j];
#pragma unroll
        for (int k = 0; k < HID; ++k) a = fmaf(g[k], w1[k * HID + j], a);
        a = (a >= 0.f) ? a : RRELU_SLOPE * a;  // rrelu (eval mode)
        z = fmaf(a, w2[j], z);
    }
    out[i] = (z >= 0.f) ? z : RRELU_SLOPE * z;
}

// ---------------------------------------------------------------------------
// Launcher
// ---------------------------------------------------------------------------
extern "C" void kernel_launch(void* const* d_in, const int* in_sizes, int n_in,
                              void* d_out, int out_size, void* d_ws, size_t ws_size,
                              hipStream_t stream) {
    const float* x       = (const float*)d_in[0];
    const int*   ei      = (const int*)d_in[1];
    const int*   batch   = (const int*)d_in[2];
    const float* W[5]    = { (const float*)d_in[3], (const float*)d_in[5],
                             (const float*)d_in[7], (const float*)d_in[9],
                             (const float*)d_in[11] };
    const float* B[5]    = { (const float*)d_in[4], (const float*)d_in[6],
                             (const float*)d_in[8], (const float*)d_in[10],
                             (const float*)d_in[12] };
    const float* lin1_w  = (const float*)d_in[13];
    const float* lin1_b  = (const float*)d_in[14];
    const float* lin2_w  = (const float*)d_in[15];
    const float* lin2_b  = (const float*)d_in[16];

    const int N = in_sizes[0] / 4;       // nodes (x is [N,4])
    const int E = in_sizes[1] / 2;       // edges (edge_index is [2,E])
    const int G = out_size;              // graphs (output is [G,1])
    const int* erow = ei;
    const int* ecol = ei + E;

    // Workspace layout (floats), 16-float aligned segments
    float* ws   = (float*)d_ws;
    size_t o    = 0;
    float* dis  = ws + o; o += ((size_t)N + 15) & ~15ull;
    float* xw   = ws + o; o += ((size_t)N * HID + 15) & ~15ull;
    float* acc  = ws + o; o += ((size_t)N * HID + 15) & ~15ull;
    float* gsum = ws + o; o += ((size_t)G * HID + 15) & ~15ull;
    float* gcnt = ws + o;
    (void)ws_size; (void)n_in;

    const int T = 256;
    const int nb_N  = (N + T - 1) / T;
    const int nb_E  = (E + T - 1) / T;
    const int tiles = (N + 15) / 16;                 // one 16-row tile per wave
    const int nb_W  = (tiles + (T / 32) - 1) / (T / 32);

    // --- degree / symmetric normalization ---
    k_init_deg<<<nb_N, T, 0, stream>>>(dis, N);
    k_deg_accum<<<nb_E, T, 0, stream>>>(ecol, dis, E);
    k_deg_to_dis<<<nb_N, T, 0, stream>>>(dis, N);

    // --- layer 1: input x [N,4], K=4 -> single WMMA per tile ---
    k_dense_wmma<<<nb_W, T, 0, stream>>>(x, 4, 4, W[0], xw, N, 0);
    k_init_acc<<<nb_N, T, 0, stream>>>(dis, xw, B[0], acc, N);
    k_edge_scatter<<<nb_E, T, 0, stream>>>(erow, ecol, dis, xw, acc, E);

    // --- layers 2..5: relu fused into dense load ---
    for (int l = 1; l < 5; ++l) {
        k_dense_wmma<<<nb_W, T, 0, stream>>>(acc, HID, HID, W[l], xw, N, 1);
        k_init_acc<<<nb_N, T, 0, stream>>>(dis, xw, B[l], acc, N);
        k_edge_scatter<<<nb_E, T, 0, stream>>>(erow, ecol, dis, xw, acc, E);
    }

    // --- global mean pool (relu fused) + MLP head ---
    k_zero<<<(G * (HID + 1) + T - 1) / T, T, 0, stream>>>(gsum, G * HID + ((gcnt - gsum) - G * HID) + G);
    k_zero<<<(G + T - 1) / T, T, 0, stream>>>(gcnt, G);
    k_pool<<<nb_N, T, 0, stream>>>(acc, batch, gsum, gcnt, N);
    k_mlp<<<(G + T - 1) / T, T, 0, stream>>>(gsum, gcnt, lin1_w, lin1_b,
                                             lin2_w, lin2_b, (float*)d_out, G);
}